// Voxelization_12816182411898
// MI455X (gfx1250) — compile-verified
//
#include <hip/hip_runtime.h>
#include <hip/hip_bf16.h>
#include <stdint.h>

#define VRES   128
#define VRES3  (VRES * VRES * VRES)
#define BATCH  2
#define NVERT  7700
#define NSURF_ 6890
#define NTET   30000
#define TETW   5
#define SPLATR 3
#define SIGMA_ 0.05f
#define EPSW   0.001f

// native clang vector (accepted by __builtin_nontemporal_load/store),
// layout-identical to HIP float4
typedef __attribute__((ext_vector_type(4))) float f32x4;

// ---------------------------------------------------------------------------
// Pass 2 (listed first so its asm shows in the disasm snippet):
// Gaussian splat. One block per (batch, surface-vertex).
// The 6 floats (pos.xyz, code.xyz) are staged to LDS with the CDNA5 async
// global->LDS path (global_load_async_to_lds_b32 + s_wait_asynccnt), then
// 343 lanes each splat one voxel of the 7x7x7 window with 4 atomic f32 adds
// into a single 16B-aligned float4 (one cacheline per splat-voxel, resolved
// in the 192MB L2 where the whole 64MB accumulator lives).
// ---------------------------------------------------------------------------
__global__ __launch_bounds__(352) void splat_kernel(const float* __restrict__ verts,
                                                    const float* __restrict__ code,
                                                    float4*      __restrict__ accum) {
    __shared__ float sdat[8];

    const int bv = blockIdx.x;
    const int b  = bv / NSURF_;
    const int v  = bv - b * NSURF_;
    const int t  = threadIdx.x;

    if (t < 6) {
        const float* src = (t < 3)
            ? (verts + ((size_t)b * NVERT  + v) * 3 + t)
            : (code  + ((size_t)b * NSURF_ + v) * 3 + (t - 3));
        // low 32 bits of a generic LDS pointer == wave-relative LDS offset
        uint32_t lds_off = (uint32_t)(uintptr_t)(&sdat[t]);
        asm volatile("global_load_async_to_lds_b32 %0, %1, off"
                     :
                     : "v"(lds_off), "v"(src)
                     : "memory");
    }
    asm volatile("s_wait_asynccnt 0x0" ::: "memory");   // no-op for non-issuing waves
    __syncthreads();

    const float vx = sdat[0], vy = sdat[1], vz = sdat[2];
    const float cdx = sdat[3], cdy = sdat[4], cdz = sdat[5];

    const int b0 = (int)floorf(vx * (float)VRES);
    const int b1 = (int)floorf(vy * (float)VRES);
    const int b2 = (int)floorf(vz * (float)VRES);

    const int W = 2 * SPLATR + 1;                       // 7
    if (t < W * W * W) {
        const int s0 = b0 + (t % W) - SPLATR;           // coord 0 fastest in memory
        const int s1 = b1 + ((t / W) % W) - SPLATR;
        const int s2 = b2 + (t / (W * W)) - SPLATR;

        const bool ok = (s0 >= 0) & (s0 < VRES) & (s1 >= 0) & (s1 < VRES) &
                        (s2 >= 0) & (s2 < VRES);
        if (ok) {
            const float px = ((float)s0 + 0.5f) * (1.0f / (float)VRES) - vx;
            const float py = ((float)s1 + 0.5f) * (1.0f / (float)VRES) - vy;
            const float pz = ((float)s2 + 0.5f) * (1.0f / (float)VRES) - vz;
            const float d2 = px * px + py * py + pz * pz;
            const float w  = __expf(-d2 * (1.0f / (2.0f * SIGMA_ * SIGMA_)));

            const size_t lin = (((size_t)b * VRES + s2) * VRES + s1) * VRES + s0;
            float4* p = accum + lin;
            atomicAdd(&p->x, w);          // wsum
            atomicAdd(&p->y, w * cdx);    // sem.x
            atomicAdd(&p->z, w * cdy);    // sem.y
            atomicAdd(&p->w, w * cdz);    // sem.z
        }
    }
}

// ---------------------------------------------------------------------------
// Pass 0: initialize workspace (accum = {EPS_W,0,0,0}, occ = 0)
// ---------------------------------------------------------------------------
__global__ __launch_bounds__(256) void init_ws_kernel(float4* __restrict__ accum,
                                                      float*  __restrict__ occ,
                                                      int n) {
    int i = blockIdx.x * blockDim.x + threadIdx.x;
    if (i < n) {
        accum[i] = make_float4(EPSW, 0.0f, 0.0f, 0.0f);
        occ[i]   = 0.0f;
    }
}

// ---------------------------------------------------------------------------
// Pass 1: tetrahedra occupancy. One block per (batch, tet); 125 lanes test
// one voxel each of the 5x5x5 window. Tet data is block-uniform -> scalar
// loads. Writers only ever write 1.0, so plain stores replace atomic-max.
// FP expression order for l1/l2/l3 matches the reference exactly (c23
// precomputed once; cross-then-dot per voxel for l2/l3).
// ---------------------------------------------------------------------------
__global__ __launch_bounds__(128) void tet_kernel(const float* __restrict__ verts,
                                                  const int*   __restrict__ tets,
                                                  float*       __restrict__ occ) {
    const int bt = blockIdx.x;
    const int b  = bt / NTET;
    const int t  = bt - b * NTET;

    const int4 ti = *(const int4*)(tets + 4 * t);          // uniform -> s_load_b128
    const float* vb = verts + (size_t)b * NVERT * 3;

    const float ax = vb[ti.x * 3 + 0], ay = vb[ti.x * 3 + 1], az = vb[ti.x * 3 + 2];
    const float bx = vb[ti.y * 3 + 0], by = vb[ti.y * 3 + 1], bz = vb[ti.y * 3 + 2];
    const float cx = vb[ti.z * 3 + 0], cy = vb[ti.z * 3 + 1], cz = vb[ti.z * 3 + 2];
    const float dx = vb[ti.w * 3 + 0], dy = vb[ti.w * 3 + 1], dz = vb[ti.w * 3 + 2];

    const float e1x = bx - ax, e1y = by - ay, e1z = bz - az;
    const float e2x = cx - ax, e2y = cy - ay, e2z = cz - az;
    const float e3x = dx - ax, e3y = dy - ay, e3z = dz - az;

    // c23 = cross(e2, e3)  (precomputed once, same as reference)
    const float c23x = e2y * e3z - e2z * e3y;
    const float c23y = e2z * e3x - e2x * e3z;
    const float c23z = e2x * e3y - e2y * e3x;

    const float vol6  = e1x * c23x + e1y * c23y + e1z * c23z;
    const bool  volok = fabsf(vol6) > 1e-12f;
    const float iv    = 1.0f / (volok ? vol6 : 1.0f);

    const float mnx = fminf(fminf(ax, bx), fminf(cx, dx));
    const float mny = fminf(fminf(ay, by), fminf(cy, dy));
    const float mnz = fminf(fminf(az, bz), fminf(cz, dz));
    const int an0 = (int)floorf(mnx * (float)VRES);
    const int an1 = (int)floorf(mny * (float)VRES);
    const int an2 = (int)floorf(mnz * (float)VRES);

    const int lane = threadIdx.x;
    if (lane < TETW * TETW * TETW) {
        const int v0 = an0 + (lane % TETW);             // coord 0 fastest in memory
        const int v1 = an1 + ((lane / TETW) % TETW);
        const int v2 = an2 + (lane / (TETW * TETW));

        const float px = ((float)v0 + 0.5f) * (1.0f / (float)VRES) - ax;
        const float py = ((float)v1 + 0.5f) * (1.0f / (float)VRES) - ay;
        const float pz = ((float)v2 + 0.5f) * (1.0f / (float)VRES) - az;

        const float l1 = (px * c23x + py * c23y + pz * c23z) * iv;

        // l2 = dot(cross(p, e3), e1) * iv
        const float q2x = py * e3z - pz * e3y;
        const float q2y = pz * e3x - px * e3z;
        const float q2z = px * e3y - py * e3x;
        const float l2 = (q2x * e1x + q2y * e1y + q2z * e1z) * iv;

        // l3 = dot(cross(e2, p), e1) * iv
        const float q3x = e2y * pz - e2z * py;
        const float q3y = e2z * px - e2x * pz;
        const float q3z = e2x * py - e2y * px;
        const float l3 = (q3x * e1x + q3y * e1y + q3z * e1z) * iv;

        const bool inside = (l1 >= 0.0f) & (l2 >= 0.0f) & (l3 >= 0.0f) &
                            ((l1 + l2 + l3) <= 1.0f) & volok;
        const bool inbox = (v0 >= 0) & (v0 < VRES) & (v1 >= 0) & (v1 < VRES) &
                           (v2 >= 0) & (v2 < VRES);
        if (inside & inbox) {
            const size_t lin = (((size_t)b * VRES + v2) * VRES + v1) * VRES + v0;
            occ[lin] = 1.0f;   // max-scatter of constant 1 -> plain store
        }
    }
}

// ---------------------------------------------------------------------------
// Pass 3: normalize + channel-first transpose.
// out[b, c, z, y, x] = sem_c / wsum * occ
// Accumulator is read exactly once and the output written exactly once ->
// non-temporal hints keep this streaming pass out of L2's way.
// ---------------------------------------------------------------------------
__global__ __launch_bounds__(256) void finalize_kernel(const f32x4* __restrict__ accum,
                                                       const float* __restrict__ occ,
                                                       float*       __restrict__ out) {
    const int i = blockIdx.x * blockDim.x + threadIdx.x;
    if (i >= BATCH * VRES3) return;
    const int b = i / VRES3;
    const int s = i - b * VRES3;

    const f32x4 a = __builtin_nontemporal_load(accum + i);
    const float o = __builtin_nontemporal_load(occ + i);
    const float invw = o / a.x;   // (sem/wsum)*occ

    const size_t base = (size_t)b * 3 * VRES3 + s;
    __builtin_nontemporal_store(a.y * invw, out + base + 0 * (size_t)VRES3);
    __builtin_nontemporal_store(a.z * invw, out + base + 1 * (size_t)VRES3);
    __builtin_nontemporal_store(a.w * invw, out + base + 2 * (size_t)VRES3);
}

// ---------------------------------------------------------------------------
extern "C" void kernel_launch(void* const* d_in, const int* in_sizes, int n_in,
                              void* d_out, int out_size, void* d_ws, size_t ws_size,
                              hipStream_t stream) {
    const float* smpl_vertices = (const float*)d_in[0];  // (B, NV, 3)
    const float* vertex_code   = (const float*)d_in[1];  // (B, NSURF, 3)
    const int*   face_indices  = (const int*)d_in[2];    // (NF, 3) -- dead in reference
    const int*   tet_indices   = (const int*)d_in[3];    // (NT, 4)
    (void)face_indices; (void)in_sizes; (void)n_in; (void)out_size; (void)ws_size;

    float* outp = (float*)d_out;

    // workspace layout: [ accum float4 x B*R3 | occ float x B*R3 ]  = 80 MB
    float4* accum = (float4*)d_ws;
    float*  occ   = (float*)((char*)d_ws + (size_t)BATCH * VRES3 * sizeof(float4));

    const int n = BATCH * VRES3;

    init_ws_kernel<<<(n + 255) / 256, 256, 0, stream>>>(accum, occ, n);
    tet_kernel<<<BATCH * NTET, 128, 0, stream>>>(smpl_vertices, tet_indices, occ);
    splat_kernel<<<BATCH * NSURF_, 352, 0, stream>>>(smpl_vertices, vertex_code, accum);
    finalize_kernel<<<(n + 255) / 256, 256, 0, stream>>>((const f32x4*)accum, occ, outp);
}